// WrappedOriginalGCN_36472862278098
// MI455X (gfx1250) — compile-verified
//
#include <hip/hip_runtime.h>
#include <cstdint>

#define NNODES 8192
#define NFEAT  128
#define NHID   256
#define NCLASS 16

typedef __attribute__((ext_vector_type(2))) float v2f;
typedef __attribute__((ext_vector_type(8))) float v8f;

// ---------------------------------------------------------------- degree ----
__global__ void zero_deg_kernel(float* __restrict__ deg) {
    int i = blockIdx.x * blockDim.x + threadIdx.x;
    if (i < NNODES) deg[i] = 0.0f;
}

__global__ void degree_kernel(const long long* __restrict__ ei,
                              const float* __restrict__ ew,
                              float* __restrict__ deg, int E) {
    int e = blockIdx.x * blockDim.x + threadIdx.x;
    if (e < E) {
        int row = (int)ei[e];
        atomicAdd(&deg[row], ew[e]);
    }
}

__global__ void dinv_kernel(float* __restrict__ deg) {
    int i = blockIdx.x * blockDim.x + threadIdx.x;
    if (i < NNODES) {
        float d = deg[i] + 1.0f;           // +1 self loop; always > 0
        deg[i] = rsqrtf(d);                // deg buffer becomes dinv
    }
}

// ------------------------------------------------- GEMM1: V = dinv*(X@W1) ---
// One wave per 16x16 output tile. M=8192 (512 tiles), N=256 (16 tiles), K=128.
__global__ __launch_bounds__(128) void gemm1_wmma_kernel(
        const float* __restrict__ X, const float* __restrict__ W1,
        const float* __restrict__ dinv, float* __restrict__ V) {
    int wave  = (blockIdx.x * blockDim.x + threadIdx.x) >> 5;
    int lane  = threadIdx.x & 31;
    int tileM = wave >> 4;           // 0..511
    int tileN = wave & 15;           // 0..15
    int row0 = tileM * 16, col0 = tileN * 16;
    int m    = lane & 15;            // row (A) / col (B,C)
    int kk   = (lane >> 4) * 2;      // K sub-offset per half-wave
    v8f c = {};
    for (int k0 = 0; k0 < NFEAT; k0 += 4) {
        v2f a, b;
        a.x = X[(row0 + m) * NFEAT + k0 + kk];
        a.y = X[(row0 + m) * NFEAT + k0 + kk + 1];
        b.x = W1[(k0 + kk) * NHID + col0 + m];
        b.y = W1[(k0 + kk + 1) * NHID + col0 + m];
        c = __builtin_amdgcn_wmma_f32_16x16x4_f32(false, a, false, b,
                                                  (short)0, c, false, false);
    }
    int half = lane >> 4;
#pragma unroll
    for (int r = 0; r < 8; ++r) {
        int row = row0 + r + 8 * half;
        V[row * NHID + col0 + m] = dinv[row] * c[r];
    }
}

// ----------------------------------------------------- sparse aggregation ---
__global__ void self_copy1_kernel(const float* __restrict__ V,
                                  float* __restrict__ agg1) {
    int i = blockIdx.x * blockDim.x + threadIdx.x;
    if (i < NNODES * NHID) agg1[i] = V[i];     // identity (self-loop) term
}

// One block per edge, 256 threads = 256 channels.
__global__ __launch_bounds__(256) void scatter1_kernel(
        const long long* __restrict__ ei, const float* __restrict__ ew,
        const float* __restrict__ V, float* __restrict__ agg1, int E) {
    int e = blockIdx.x;
    int c = threadIdx.x;
    int row = (int)ei[e];
    int col = (int)ei[(long long)E + e];
    float w = ew[e];
    atomicAdd(&agg1[row * NHID + c], w * V[col * NHID + c]);
}

__global__ void relu_kernel(const float* __restrict__ agg1,
                            const float* __restrict__ dinv,
                            const float* __restrict__ b1,
                            float* __restrict__ H) {
    int idx = blockIdx.x * blockDim.x + threadIdx.x;
    if (idx < NNODES * NHID) {
        int i = idx >> 8;          // / NHID
        int c = idx & (NHID - 1);
        float v = dinv[i] * agg1[idx] + b1[c];
        H[idx] = v > 0.0f ? v : 0.0f;
    }
}

// ------------------------------------------------ GEMM2: Vh = dinv*(H@W2) ---
// M=8192 (512 tiles), N=16 (1 tile), K=256.
__global__ __launch_bounds__(128) void gemm2_wmma_kernel(
        const float* __restrict__ H, const float* __restrict__ W2,
        const float* __restrict__ dinv, float* __restrict__ Vh) {
    int wave = (blockIdx.x * blockDim.x + threadIdx.x) >> 5;
    int lane = threadIdx.x & 31;
    int row0 = wave * 16;
    int m    = lane & 15;
    int kk   = (lane >> 4) * 2;
    v8f c = {};
    for (int k0 = 0; k0 < NHID; k0 += 4) {
        v2f a, b;
        a.x = H[(row0 + m) * NHID + k0 + kk];
        a.y = H[(row0 + m) * NHID + k0 + kk + 1];
        b.x = W2[(k0 + kk) * NCLASS + m];
        b.y = W2[(k0 + kk + 1) * NCLASS + m];
        c = __builtin_amdgcn_wmma_f32_16x16x4_f32(false, a, false, b,
                                                  (short)0, c, false, false);
    }
    int half = lane >> 4;
#pragma unroll
    for (int r = 0; r < 8; ++r) {
        int row = row0 + r + 8 * half;
        Vh[row * NCLASS + m] = dinv[row] * c[r];
    }
}

__global__ void self_copy2_kernel(const float* __restrict__ Vh,
                                  float* __restrict__ agg2) {
    int i = blockIdx.x * blockDim.x + threadIdx.x;
    if (i < NNODES * NCLASS) agg2[i] = Vh[i];
}

// Each thread handles one (edge, class) pair: idx>>4 = edge, idx&15 = class.
__global__ __launch_bounds__(256) void scatter2_kernel(
        const long long* __restrict__ ei, const float* __restrict__ ew,
        const float* __restrict__ Vh, float* __restrict__ agg2, int E) {
    int idx = blockIdx.x * blockDim.x + threadIdx.x;
    int e = idx >> 4;
    int c = idx & 15;
    if (e < E) {
        int row = (int)ei[e];
        int col = (int)ei[(long long)E + e];
        float w = ew[e];
        atomicAdd(&agg2[row * NCLASS + c], w * Vh[col * NCLASS + c]);
    }
}

// --------------------------------------------- bias + log_softmax (16 cls) --
__global__ void logsoftmax_kernel(const float* __restrict__ agg2,
                                  const float* __restrict__ dinv,
                                  const float* __restrict__ b2,
                                  float* __restrict__ out) {
    int i = blockIdx.x * blockDim.x + threadIdx.x;
    if (i >= NNODES) return;
    float di = dinv[i];
    float o[NCLASS];
    float mx = -3.0e38f;
#pragma unroll
    for (int c = 0; c < NCLASS; ++c) {
        o[c] = di * agg2[i * NCLASS + c] + b2[c];
        mx = fmaxf(mx, o[c]);
    }
    float s = 0.0f;
#pragma unroll
    for (int c = 0; c < NCLASS; ++c) s += expf(o[c] - mx);
    float lse = mx + logf(s);
#pragma unroll
    for (int c = 0; c < NCLASS; ++c) out[i * NCLASS + c] = o[c] - lse;
}

// ---------------------------------------------------------------- launch ----
extern "C" void kernel_launch(void* const* d_in, const int* in_sizes, int n_in,
                              void* d_out, int out_size, void* d_ws, size_t ws_size,
                              hipStream_t stream) {
    const float*     x  = (const float*)d_in[0];
    const long long* ei = (const long long*)d_in[1];   // int64 (2, E)
    const float*     ew = (const float*)d_in[2];
    const float*     W1 = (const float*)d_in[3];
    const float*     b1 = (const float*)d_in[4];
    const float*     W2 = (const float*)d_in[5];
    const float*     b2 = (const float*)d_in[6];
    float* out = (float*)d_out;
    const int E = in_sizes[1] / 2;

    // workspace layout (floats)
    float* ws   = (float*)d_ws;
    float* dinv = ws;                                  // 8192   (deg -> dinv)
    float* V    = dinv + NNODES;                       // 8192*256  (also H)
    float* agg1 = V + (size_t)NNODES * NHID;           // 8192*256
    float* Vh   = agg1 + (size_t)NNODES * NHID;        // 8192*16
    float* agg2 = Vh + (size_t)NNODES * NCLASS;        // 8192*16

    // 1) degree + dinv
    zero_deg_kernel<<<NNODES / 256, 256, 0, stream>>>(dinv);
    degree_kernel<<<(E + 255) / 256, 256, 0, stream>>>(ei, ew, dinv, E);
    dinv_kernel<<<NNODES / 256, 256, 0, stream>>>(dinv);

    // 2) V = dinv * (X @ W1)   [WMMA f32]
    gemm1_wmma_kernel<<<(NNODES / 16) * (NHID / 16) / 4, 128, 0, stream>>>(x, W1, dinv, V);

    // 3) agg1 = V + scatter(w * V[col])
    self_copy1_kernel<<<NNODES * NHID / 256, 256, 0, stream>>>(V, agg1);
    scatter1_kernel<<<E, NHID, 0, stream>>>(ei, ew, V, agg1, E);

    // 4) H = relu(dinv * agg1 + b1)  (overwrites V)
    relu_kernel<<<NNODES * NHID / 256, 256, 0, stream>>>(agg1, dinv, b1, V);

    // 5) Vh = dinv * (H @ W2)   [WMMA f32]
    gemm2_wmma_kernel<<<(NNODES / 16) / 4, 128, 0, stream>>>(V, W2, dinv, Vh);

    // 6) agg2 = Vh + scatter(w * Vh[col])
    self_copy2_kernel<<<NNODES * NCLASS / 256, 256, 0, stream>>>(Vh, agg2);
    scatter2_kernel<<<(E * NCLASS) / 256, 256, 0, stream>>>(ei, ew, Vh, agg2, E);

    // 7) out = log_softmax(dinv * agg2 + b2)
    logsoftmax_kernel<<<NNODES / 256, 256, 0, stream>>>(agg2, dinv, b2, out);
}